// CrossLevelAttention_71545565216848
// MI455X (gfx1250) — compile-verified
//
#include <hip/hip_runtime.h>
#include <math.h>

// ---------------------------------------------------------------------------
// CrossLevelAttention for MI455X (gfx1250, wave32, WMMA f32 16x16x4)
//
// Big GEMMs on V_WMMA_F32_16X16X4_F32. W slab is stored transposed in LDS so
// each B operand is a single ds_load_b64 into an adjacent VGPR pair (compiler
// fuses k-step pairs into ds_load_2addr_b64 + 2 WMMAs). X tile staging uses
// the CDNA5 async global->LDS path (GLOBAL_LOAD_ASYNC_TO_LDS_B128, ASYNCcnt).
// Segment softmax/reduction uses f32 global atomics resolved in L2.
// ---------------------------------------------------------------------------

#define NC    200000
#define NT    2000
#define Hdim  256
#define NHh   8
#define HDd   32
#define SCALE 0.17677669529663687f   // 32^-0.5

typedef __attribute__((ext_vector_type(2))) float v2f;
typedef __attribute__((ext_vector_type(8))) float v8f;
typedef int v4i_ __attribute__((__vector_size__(16)));   // matches builtin param

#ifndef __has_builtin
#define __has_builtin(x) 0
#endif
#if __has_builtin(__builtin_amdgcn_global_load_async_to_lds_b128)
#define HAVE_ASYNC_LDS 1
#else
#define HAVE_ASYNC_LDS 0
#endif

#if HAVE_ASYNC_LDS
__device__ __forceinline__ void wait_async0() {
#if __has_builtin(__builtin_amdgcn_s_wait_asynccnt)
  __builtin_amdgcn_s_wait_asynccnt(0);
#else
  asm volatile("s_wait_asynccnt 0x0" ::: "memory");
#endif
}
#endif

// Xs: row-major 64x256, stride 260 (16B-aligned rows; 260%64==4 -> the 16
//     A-rows of a WMMA hit distinct banks).
// Wt: transposed slab [256 cols][32 k], stride 34 (even -> b64-aligned;
//     34c mod 64 distinct for c=0..15 -> conflict-free B loads).
#define XS_STRIDE 260
#define WT_STRIDE 34
#define GEMM_LDS_FLOATS (64 * XS_STRIDE + 256 * WT_STRIDE)

__device__ __forceinline__ void atomicMaxF(float* addr, float val) {
  int old = __float_as_int(*addr);
  while (__int_as_float(old) < val) {
    int prev = atomicCAS((int*)addr, old, __float_as_int(val));
    if (prev == old) break;
    old = prev;
  }
}

// ---------------------------------------------------------------------------
// Core: 64-row x 256-col tile of C = A[M,256] @ W[256,256], WMMA f32 16x16x4.
// Block = 128 threads = 4 waves; wave w owns rows [rowBase+16w, rowBase+16w+16).
// Accumulator layout (ISA 7.12.2): acc[n][i] = C[16w + i + 8*(lane>=16)][16n + lane%16]
// ---------------------------------------------------------------------------
__device__ __forceinline__ void gemm_core(const float* __restrict__ A,
                                          const float* __restrict__ W,
                                          int M, int rowBase,
                                          float* Xs, float* Wt, v8f* acc) {
  const int tid  = threadIdx.x;
  const int lane = tid & 31;
  const int wave = tid >> 5;
  const int half = lane >> 4;      // 0: lanes 0-15, 1: lanes 16-31
  const int col  = lane & 15;
  const int mrow = wave * 16;

  const v8f vzero = {0.f,0.f,0.f,0.f,0.f,0.f,0.f,0.f};
#pragma unroll
  for (int n = 0; n < 16; ++n) acc[n] = vzero;

  // stage X tile: 64 rows x 256 cols
#if HAVE_ASYNC_LDS
  // CDNA5 async DMA: global -> LDS without the VGPR round-trip (ASYNCcnt).
  for (int i = tid; i < 64 * 64; i += 128) {
    int r  = i >> 6;
    int c4 = (i & 63) << 2;
    int gr = rowBase + r; if (gr >= M) gr = M - 1;
    __builtin_amdgcn_global_load_async_to_lds_b128(
        (v4i_*)(A + (size_t)gr * Hdim + c4),
        (v4i_*)(Xs + r * XS_STRIDE + c4),
        0, 0);
  }
  wait_async0();
#else
  for (int i = tid; i < 64 * 64; i += 128) {
    int r  = i >> 6;
    int c4 = (i & 63) << 2;
    int gr = rowBase + r; if (gr >= M) gr = M - 1;
    const float4 v = *(const float4*)(A + (size_t)gr * Hdim + c4);
    *(float4*)(Xs + r * XS_STRIDE + c4) = v;
  }
#endif

  const float* xrow  = Xs + (mrow + col) * XS_STRIDE + 2 * half;
  const float* wbase = Wt + col * WT_STRIDE + 2 * half;

  for (int kk = 0; kk < Hdim; kk += 32) {
    __syncthreads();                      // X ready / previous slab consumed
    // stage W slab [kk,kk+32) x 256, transposed: Wt[c][k] = W[kk+k][c]
    for (int i = tid; i < 32 * 64; i += 128) {
      int r  = i >> 6;                    // k within slab
      int c4 = (i & 63) << 2;             // column
      const float4 v = *(const float4*)(W + (size_t)(kk + r) * Hdim + c4);
      Wt[(c4 + 0) * WT_STRIDE + r] = v.x;
      Wt[(c4 + 1) * WT_STRIDE + r] = v.y;
      Wt[(c4 + 2) * WT_STRIDE + r] = v.z;
      Wt[(c4 + 3) * WT_STRIDE + r] = v.w;
    }
    __syncthreads();
#pragma unroll
    for (int ks = 0; ks < 32; ks += 4) {
      // A operand (16x4): lane holds X[mrow+col][k + 2*half .. +1]
      v2f a = *(const v2f*)(xrow + kk + ks);
      // One b64 per B operand (shared base + imm offsets); the scheduler
      // fuses k-step pairs into ds_load_2addr_b64 feeding 2 WMMAs.
      v2f bArr[16];
#pragma unroll
      for (int n = 0; n < 16; ++n)
        bArr[n] = *(const v2f*)(wbase + n * (16 * WT_STRIDE) + ks);
#pragma unroll
      for (int n = 0; n < 16; ++n)
        acc[n] = __builtin_amdgcn_wmma_f32_16x16x4_f32(
            false, a, false, bArr[n], (short)0, acc[n], false, false);
    }
  }
}

// ---------------------------------------------------------------------------
// Generic C = A@W + bias (used for Q proj and the 2000-row tissue GEMMs)
// ---------------------------------------------------------------------------
__global__ __launch_bounds__(128, 1)
void gemm_bias(const float* __restrict__ A, const float* __restrict__ W,
               const float* __restrict__ bias, float* __restrict__ C, int M) {
  extern __shared__ float smem[];
  float* Xs = smem;
  float* Wt = smem + 64 * XS_STRIDE;
  v8f acc[16];
  const int rowBase = blockIdx.x * 64;
  gemm_core(A, W, M, rowBase, Xs, Wt, acc);

  const int lane = threadIdx.x & 31, wave = threadIdx.x >> 5;
  const int half = lane >> 4, col = lane & 15;
#pragma unroll
  for (int n = 0; n < 16; ++n) {
    const float bb = bias[n * 16 + col];
#pragma unroll
    for (int mi = 0; mi < 8; ++mi) {
      int gr = rowBase + wave * 16 + mi + 8 * half;
      if (gr < M) C[(size_t)gr * Hdim + n * 16 + col] = acc[n][mi] + bb;
    }
  }
}

// ---------------------------------------------------------------------------
// Pass A: K = cell @ Wk + bk (never hits HBM), per-(cell,head) score vs its
// cluster's Q, atomic segment-max into smax, scores -> ws, counts.
// ---------------------------------------------------------------------------
__global__ __launch_bounds__(128, 1)
void cell_scores(const float* __restrict__ cellF, const float* __restrict__ Wk,
                 const float* __restrict__ bk, const int* __restrict__ labels,
                 const float* __restrict__ Q, float* __restrict__ scores,
                 float* __restrict__ smax, float* __restrict__ counts, int M) {
  extern __shared__ float smem[];
  float* Xs = smem;
  float* Wt = smem + 64 * XS_STRIDE;
  int*   clu = (int*)(smem + GEMM_LDS_FLOATS);

  const int rowBase = blockIdx.x * 64;
  const int tid = threadIdx.x;
  if (tid < 64) {
    int g = rowBase + tid;
    int gc = g < M ? g : M - 1;
    clu[tid] = labels[gc];
    if (g < M) atomicAdd(&counts[labels[gc]], 1.0f);
  }

  v8f acc[16];
  gemm_core(cellF, Wk, M, rowBase, Xs, Wt, acc);   // barriers inside make clu visible

  const int lane = tid & 31, wave = tid >> 5;
  const int half = lane >> 4, col = lane & 15;
#pragma unroll
  for (int n = 0; n < 16; ++n) {                    // add k-proj bias
    const float bb = bk[n * 16 + col];
#pragma unroll
    for (int mi = 0; mi < 8; ++mi) acc[n][mi] += bb;
  }

  // score(cell, h) = SCALE * sum_d K[cell][32h+d] * Q[cluster][32h+d]
  // lanes 0-15 reduce cells mi (rows i), lanes 16-31 reduce cells mi+8.
#pragma unroll
  for (int h = 0; h < NHh; ++h) {
#pragma unroll
    for (int mi = 0; mi < 8; ++mi) {
      const int cellL = wave * 16 + mi + 8 * half;
      const int c = clu[cellL];
      const float* q = Q + (size_t)c * Hdim + h * 32;
      float p = acc[2 * h][mi] * q[col] + acc[2 * h + 1][mi] * q[16 + col];
#pragma unroll
      for (int o = 1; o < 16; o <<= 1) p += __shfl_xor(p, o, 32);
      if (col == 0) {
        int g = rowBase + cellL;
        if (g < M) {
          float s = p * SCALE;
          scores[(size_t)g * NHh + h] = s;
          atomicMaxF(&smax[c * NHh + h], s);
        }
      }
    }
  }
}

// ---------------------------------------------------------------------------
// Pass B: V = cell @ Wv + bv, ex = exp(score - smax), atomic denom + weighted
// scatter of ex*V into per-tissue accumulator (2MB, L2-resident).
// ---------------------------------------------------------------------------
__global__ __launch_bounds__(128, 1)
void cell_v_scatter(const float* __restrict__ cellF, const float* __restrict__ Wv,
                    const float* __restrict__ bv, const int* __restrict__ labels,
                    const float* __restrict__ scores, const float* __restrict__ smax,
                    float* __restrict__ denom, float* __restrict__ attended, int M) {
  extern __shared__ float smem[];
  float* Xs  = smem;
  float* Wt  = smem + 64 * XS_STRIDE;
  int*   clu = (int*)(smem + GEMM_LDS_FLOATS);
  float* Exs = smem + GEMM_LDS_FLOATS + 64;        // [64 cells][8 heads]

  const int rowBase = blockIdx.x * 64;
  const int tid = threadIdx.x;
  if (tid < 64) {
    int g = rowBase + tid;
    clu[tid] = labels[g < M ? g : M - 1];
  }

  v8f acc[16];
  gemm_core(cellF, Wv, M, rowBase, Xs, Wt, acc);

  const int lane = tid & 31, wave = tid >> 5;
  const int half = lane >> 4, col = lane & 15;
#pragma unroll
  for (int n = 0; n < 16; ++n) {                    // add v-proj bias
    const float bb = bv[n * 16 + col];
#pragma unroll
    for (int mi = 0; mi < 8; ++mi) acc[n][mi] += bb;
  }

  // exp weights + denominator (cooperative over the 64x8 tile)
  for (int i = tid; i < 64 * NHh; i += 128) {
    int cellL = i >> 3, h = i & 7;
    int g = rowBase + cellL;
    float e = 0.f;
    if (g < M) {
      int c = clu[cellL];
      e = expf(scores[(size_t)g * NHh + h] - smax[c * NHh + h]);
      atomicAdd(&denom[c * NHh + h], e);
    }
    Exs[i] = e;
  }
  __syncthreads();

  // attended[c][:] += ex * V   (fire-and-forget f32 atomics into L2)
#pragma unroll
  for (int n = 0; n < 16; ++n) {
    const int h = n >> 1;
#pragma unroll
    for (int mi = 0; mi < 8; ++mi) {
      const int cellL = wave * 16 + mi + 8 * half;
      const float e = Exs[cellL * NHh + h];
      const int c = clu[cellL];
      atomicAdd(&attended[(size_t)c * Hdim + n * 16 + col], acc[n][mi] * e);
    }
  }
}

// ---------------------------------------------------------------------------
// Small elementwise kernels
// ---------------------------------------------------------------------------
__global__ void init_ws(float* attended, float* denom, float* smax, float* counts) {
  int i = blockIdx.x * 256 + threadIdx.x;
  if (i < NT * Hdim) attended[i] = 0.f;
  if (i < NT * NHh) { denom[i] = 0.f; smax[i] = -INFINITY; }
  if (i < NT) counts[i] = 0.f;
}

__global__ void norm_attended(float* attended, const float* __restrict__ denom) {
  int i = blockIdx.x * 256 + threadIdx.x;
  if (i < NT * Hdim) {
    int t = i >> 8, h = (i & 255) >> 5;
    attended[i] = attended[i] / denom[t * NHh + h];
  }
}

__global__ void select_tissue(const float* __restrict__ tuo, const float* __restrict__ tisF,
                              const float* __restrict__ counts, float* __restrict__ tu) {
  int i = blockIdx.x * 256 + threadIdx.x;
  if (i < NT * Hdim) tu[i] = counts[i >> 8] > 0.f ? tuo[i] : tisF[i];
}

// residual + layernorm, one wave per 256-wide row; labels!=nullptr gathers add-rows
__global__ void ln_rows(const float* __restrict__ x, const float* __restrict__ add,
                        const int* __restrict__ labels, const float* __restrict__ g,
                        const float* __restrict__ b, float* __restrict__ out, int M) {
  int row = blockIdx.x * 8 + (threadIdx.x >> 5);
  int lane = threadIdx.x & 31;
  if (row >= M) return;
  const float* xr = x + (size_t)row * Hdim;
  const float* ar = add + (size_t)(labels ? labels[row] : row) * Hdim;
  float v[8];
  float s = 0.f;
#pragma unroll
  for (int j = 0; j < 8; ++j) {
    float t = xr[lane + 32 * j] + ar[lane + 32 * j];
    v[j] = t; s += t;
  }
#pragma unroll
  for (int o = 16; o > 0; o >>= 1) s += __shfl_xor(s, o, 32);
  const float mean = s * (1.f / 256.f);
  float s2 = 0.f;
#pragma unroll
  for (int j = 0; j < 8; ++j) { float d = v[j] - mean; s2 += d * d; }
#pragma unroll
  for (int o = 16; o > 0; o >>= 1) s2 += __shfl_xor(s2, o, 32);
  const float inv = rsqrtf(s2 * (1.f / 256.f) + 1e-5f);
#pragma unroll
  for (int j = 0; j < 8; ++j) {
    int cc = lane + 32 * j;
    out[(size_t)row * Hdim + cc] = (v[j] - mean) * inv * g[cc] + b[cc];
  }
}

// ---------------------------------------------------------------------------
extern "C" void kernel_launch(void* const* d_in, const int* in_sizes, int n_in,
                              void* d_out, int out_size, void* d_ws, size_t ws_size,
                              hipStream_t stream) {
  (void)in_sizes; (void)n_in; (void)out_size; (void)ws_size;
  const float* cellF  = (const float*)d_in[0];
  const float* tisF   = (const float*)d_in[1];
  const int*   labels = (const int*)d_in[2];
  const float* Wq = (const float*)d_in[4];  const float* bq = (const float*)d_in[5];
  const float* Wk = (const float*)d_in[6];  const float* bk = (const float*)d_in[7];
  const float* Wv = (const float*)d_in[8];  const float* bv = (const float*)d_in[9];
  const float* Wo = (const float*)d_in[10]; const float* bo = (const float*)d_in[11];
  const float* Wvtd = (const float*)d_in[12]; const float* bvtd = (const float*)d_in[13];
  const float* Wotd = (const float*)d_in[14]; const float* botd = (const float*)d_in[15];
  const float* lncg = (const float*)d_in[16]; const float* lncb = (const float*)d_in[17];
  const float* lntg = (const float*)d_in[18]; const float* lntb = (const float*)d_in[19];

  // workspace layout (floats): ~18.8 MB
  float* ws       = (float*)d_ws;
  float* Qw       = ws;                       // NT*H
  float* scores   = Qw + NT * Hdim;           // NC*NH
  float* smax     = scores + (size_t)NC * NHh;// NT*NH
  float* denom    = smax + NT * NHh;          // NT*NH
  float* attended = denom + NT * NHh;         // NT*H
  float* tuo      = attended + NT * Hdim;     // NT*H
  float* tu       = tuo + NT * Hdim;          // NT*H
  float* t1       = tu + NT * Hdim;           // NT*H
  float* td       = t1 + NT * Hdim;           // NT*H
  float* counts   = td + NT * Hdim;           // NT

  float* outCell = (float*)d_out;
  float* outTis  = outCell + (size_t)NC * Hdim;

  const size_t gemmSmem = GEMM_LDS_FLOATS * sizeof(float);

  init_ws<<<(NT * Hdim + 255) / 256, 256, 0, stream>>>(attended, denom, smax, counts);

  gemm_bias<<<(NT + 63) / 64, 128, gemmSmem, stream>>>(tisF, Wq, bq, Qw, NT);

  cell_scores<<<NC / 64, 128, gemmSmem + 64 * sizeof(int), stream>>>(
      cellF, Wk, bk, labels, Qw, scores, smax, counts, NC);

  cell_v_scatter<<<NC / 64, 128, gemmSmem + (64 + 64 * NHh) * sizeof(float), stream>>>(
      cellF, Wv, bv, labels, scores, smax, denom, attended, NC);

  norm_attended<<<(NT * Hdim + 255) / 256, 256, 0, stream>>>(attended, denom);
  gemm_bias<<<(NT + 63) / 64, 128, gemmSmem, stream>>>(attended, Wo, bo, tuo, NT);
  select_tissue<<<(NT * Hdim + 255) / 256, 256, 0, stream>>>(tuo, tisF, counts, tu);
  gemm_bias<<<(NT + 63) / 64, 128, gemmSmem, stream>>>(tu, Wvtd, bvtd, t1, NT);
  gemm_bias<<<(NT + 63) / 64, 128, gemmSmem, stream>>>(t1, Wotd, botd, td, NT);

  ln_rows<<<(NT + 7) / 8, 256, 0, stream>>>(tisF, tu, nullptr, lntg, lntb, outTis, NT);
  ln_rows<<<(NC + 7) / 8, 256, 0, stream>>>(cellF, td, labels, lncg, lncb, outCell, NC);
}